// GCN_Classifier_81570018885760
// MI455X (gfx1250) — compile-verified
//
#include <hip/hip_runtime.h>
#include <hip/hip_bf16.h>
#include <math.h>
#include <stdint.h>

// fp32 WMMA fragments (wave32): A 16x4 f32 -> 2 VGPR, C/D 16x16 f32 -> 8 VGPR
typedef __attribute__((ext_vector_type(2))) float v2f;
typedef __attribute__((ext_vector_type(8))) float v8f;

#define D_IN   128
#define D_OUT  40
#define BN_EPS 1e-5f

// --------------------------------------------------------------------------
// gfx1250 async global->LDS copy (ASYNCcnt-tracked, no VGPR data movement).
// Probe round 2 revealed the builtin's signature: it takes pointers to
// int __vector(4) in the global (AS1 / __device__) and LDS (AS3) address
// spaces. Cast through uintptr_t (flat LDS addr low 32 bits == LDS offset).
// --------------------------------------------------------------------------
#if defined(__HIP_DEVICE_COMPILE__) &&                                        \
    __has_builtin(__builtin_amdgcn_global_load_async_to_lds_b128) &&          \
    __has_builtin(__builtin_amdgcn_s_wait_asynccnt)
#define USE_ASYNC_LDS 1
typedef int v4i_vs __attribute__((__vector_size__(4 * sizeof(int))));
typedef __attribute__((address_space(1))) v4i_vs* g_v4i_ptr;
typedef __attribute__((address_space(3))) v4i_vs* l_v4i_ptr;
static __device__ __forceinline__ void async_g2l_b128(const void* g, void* l) {
  __builtin_amdgcn_global_load_async_to_lds_b128(
      (g_v4i_ptr)(uintptr_t)g, (l_v4i_ptr)(uint32_t)(uintptr_t)l,
      /*offset=*/0, /*cpol=*/0);
}
#else
#define USE_ASYNC_LDS 0
#endif

// ---------------------------------------------------------------------------
// SpMM scatter: one wave per edge; lane l handles features 4l..4l+3.
// z[row] += val * x[col]   (z pre-zeroed via hipMemsetAsync)
// ---------------------------------------------------------------------------
__global__ __launch_bounds__(256) void spmm_scatter(
    const int* __restrict__ rows, const int* __restrict__ cols,
    const float* __restrict__ vals, const float* __restrict__ X,
    float* __restrict__ Z, int E) {
  int e = blockIdx.x * 8 + (threadIdx.x >> 5);
  if (e >= E) return;
  int lane = threadIdx.x & 31;
  int r = rows[e];
  int c = cols[e];
  float v = vals[e];
  const float4 xv = reinterpret_cast<const float4*>(X + (size_t)c * D_IN)[lane];
  float* zp = Z + (size_t)r * D_IN + lane * 4;
  atomicAdd(zp + 0, v * xv.x);
  atomicAdd(zp + 1, v * xv.y);
  atomicAdd(zp + 2, v * xv.z);
  atomicAdd(zp + 3, v * xv.w);
}

// ---------------------------------------------------------------------------
// GEMM1: H[50000,128] = Z[50000,128] @ W1[128,128] + b1, fp32 WMMA.
// One block = 16-row tile; A staged into LDS (async global->LDS on CDNA5);
// 8 waves each own one 16-col tile of W1.
// LDS layout As[m][k] with 132-float row stride: async-copy compatible
// (contiguous 16B chunks) AND bank-conflict-free WMMA reads
// (lanes hit banks (4m + k') mod 64; half-waves disjoint mod 4).
// ---------------------------------------------------------------------------
__global__ __launch_bounds__(256) void gemm1_bias(
    const float* __restrict__ A, const float* __restrict__ W,
    const float* __restrict__ bias, float* __restrict__ H) {
  __shared__ float As[16][132];  // 8448 B
  const int t = threadIdx.x;
  const int m0 = blockIdx.x * 16;

  // Cooperative staging: 16 rows x 128 cols = 512 float4 chunks.
#pragma unroll
  for (int i = t; i < 512; i += 256) {
    int m = i >> 5;           // row within tile
    int kk = (i & 31) * 4;    // starting k
    const float* gp = A + (size_t)(m0 + m) * D_IN + kk;
#if USE_ASYNC_LDS
    async_g2l_b128(gp, &As[m][kk]);
#else
    *reinterpret_cast<float4*>(&As[m][kk]) =
        *reinterpret_cast<const float4*>(gp);
#endif
  }
#if USE_ASYNC_LDS
  __builtin_amdgcn_s_wait_asynccnt(0);
#endif
  __syncthreads();

  const int wave = t >> 5;
  const int lane = t & 31;
  const int n0 = wave * 16;    // 8 waves cover 128 output cols
  const int mr = lane & 15;    // M index (A) / N index (B,C)
  const int kh = lane >> 4;    // K-half select

  v8f acc = {};
#pragma unroll 4
  for (int k = 0; k < D_IN; k += 4) {
    v2f a, b;
    a.x = As[mr][k + 2 * kh + 0];
    a.y = As[mr][k + 2 * kh + 1];
    b.x = W[(k + 2 * kh + 0) * D_IN + n0 + mr];
    b.y = W[(k + 2 * kh + 1) * D_IN + n0 + mr];
    acc = __builtin_amdgcn_wmma_f32_16x16x4_f32(
        false, a, false, b, (short)0, acc, false, false);
  }

  const float bb = bias[n0 + mr];
#pragma unroll
  for (int j = 0; j < 8; ++j) {
    int row = m0 + j + 8 * kh;
    H[(size_t)row * D_IN + n0 + mr] = acc[j] + bb;
  }
}

// ---------------------------------------------------------------------------
// Column sums / sums-of-squares for BatchNorm (128 threads = 128 columns).
// ---------------------------------------------------------------------------
__global__ __launch_bounds__(128) void col_stats(
    const float* __restrict__ H, float* __restrict__ sums, int N, int rowsPerBlk) {
  int c = threadIdx.x;
  int r0 = blockIdx.x * rowsPerBlk;
  int r1 = min(r0 + rowsPerBlk, N);
  float s = 0.f, s2 = 0.f;
  for (int r = r0; r < r1; ++r) {
    float v = H[(size_t)r * D_IN + c];
    s += v;
    s2 += v * v;
  }
  atomicAdd(&sums[c], s);
  atomicAdd(&sums[D_IN + c], s2);
}

// Fold gamma/beta/mean/var into per-column scale & shift.
__global__ __launch_bounds__(128) void bn_params(
    const float* __restrict__ sums, const float* __restrict__ gamma,
    const float* __restrict__ beta, float* __restrict__ prm, float invN) {
  int c = threadIdx.x;
  float mean = sums[c] * invN;
  float var = sums[D_IN + c] * invN - mean * mean;  // biased variance
  float sc = gamma[c] * rsqrtf(var + BN_EPS);
  prm[c] = sc;
  prm[D_IN + c] = beta[c] - mean * sc;
}

// Fused BN-apply + ReLU, float4 vectorized, in place on H.
__global__ __launch_bounds__(256) void bn_relu(
    float* __restrict__ H, const float* __restrict__ prm, int n4) {
  int i = blockIdx.x * 256 + threadIdx.x;
  if (i >= n4) return;
  float4 v = reinterpret_cast<float4*>(H)[i];
  int c = (i & 31) * 4;  // column of first element (row length 128 = 32 float4)
  float4 r;
  r.x = fmaxf(v.x * prm[c + 0] + prm[D_IN + c + 0], 0.f);
  r.y = fmaxf(v.y * prm[c + 1] + prm[D_IN + c + 1], 0.f);
  r.z = fmaxf(v.z * prm[c + 2] + prm[D_IN + c + 2], 0.f);
  r.w = fmaxf(v.w * prm[c + 3] + prm[D_IN + c + 3], 0.f);
  reinterpret_cast<float4*>(H)[i] = r;
}

// ---------------------------------------------------------------------------
// GEMM2 + fused log-softmax: out[50000,40] = logsoftmax(Z @ Wc + bc, axis=1).
// Each wave: 16 rows x 3 n-tiles (40 cols padded to 48, OOB cols zeroed).
// Tile parked in LDS, then lanes 0..15 do the 40-wide row reduction.
// ---------------------------------------------------------------------------
__global__ __launch_bounds__(256) void gemm2_lsm(
    const float* __restrict__ Z, const float* __restrict__ Wc,
    const float* __restrict__ bc, float* __restrict__ out, int N) {
  __shared__ float tile[8][16][48];  // 24 KB
  const int wave = threadIdx.x >> 5;
  const int lane = threadIdx.x & 31;
  const int m0 = blockIdx.x * 128 + wave * 16;
  const int mr = lane & 15;
  const int kh = lane >> 4;

  int arow = m0 + mr;
  if (arow >= N) arow = N - 1;  // clamp loads; stores are guarded below

  v8f acc[3] = {};
#pragma unroll 2
  for (int k = 0; k < D_IN; k += 4) {
    v2f a;
    a.x = Z[(size_t)arow * D_IN + k + 2 * kh + 0];
    a.y = Z[(size_t)arow * D_IN + k + 2 * kh + 1];
#pragma unroll
    for (int tt = 0; tt < 3; ++tt) {
      int col = tt * 16 + mr;
      v2f b;
      b.x = (col < D_OUT) ? Wc[(k + 2 * kh + 0) * D_OUT + col] : 0.f;
      b.y = (col < D_OUT) ? Wc[(k + 2 * kh + 1) * D_OUT + col] : 0.f;
      acc[tt] = __builtin_amdgcn_wmma_f32_16x16x4_f32(
          false, a, false, b, (short)0, acc[tt], false, false);
    }
  }

#pragma unroll
  for (int tt = 0; tt < 3; ++tt) {
    int col = tt * 16 + mr;
    float bb = (col < D_OUT) ? bc[col] : 0.f;
#pragma unroll
    for (int j = 0; j < 8; ++j)
      tile[wave][j + 8 * kh][col] = acc[tt][j] + bb;
  }
  __syncthreads();

  if (lane < 16) {
    int row = m0 + lane;
    if (row < N) {
      float mx = -INFINITY;
#pragma unroll
      for (int c = 0; c < D_OUT; ++c) mx = fmaxf(mx, tile[wave][lane][c]);
      float s = 0.f;
#pragma unroll
      for (int c = 0; c < D_OUT; ++c) s += __expf(tile[wave][lane][c] - mx);
      float lse = mx + __logf(s);
#pragma unroll
      for (int c = 0; c < D_OUT; ++c)
        out[(size_t)row * D_OUT + c] = tile[wave][lane][c] - lse;
    }
  }
}

// ---------------------------------------------------------------------------
extern "C" void kernel_launch(void* const* d_in, const int* in_sizes, int n_in,
                              void* d_out, int out_size, void* d_ws, size_t ws_size,
                              hipStream_t stream) {
  const float* x      = (const float*)d_in[0];
  const int*   rows   = (const int*)d_in[1];
  const int*   cols   = (const int*)d_in[2];
  const float* vals   = (const float*)d_in[3];
  const float* W1     = (const float*)d_in[4];
  const float* b1     = (const float*)d_in[5];
  const float* gamma1 = (const float*)d_in[6];
  const float* beta1  = (const float*)d_in[7];
  const float* Wc     = (const float*)d_in[8];
  const float* bc     = (const float*)d_in[9];
  float* out = (float*)d_out;

  const int N = in_sizes[0] / D_IN;  // 50000
  const int E = in_sizes[1];         // 800000

  const size_t featBytes = (size_t)N * D_IN * sizeof(float);
  char* ws = (char*)d_ws;
  float* z    = (float*)(ws);                    // [N,128] spmm output (reused)
  float* h    = (float*)(ws + featBytes);        // [N,128] fc1 / activations
  float* sums = (float*)(ws + 2 * featBytes);    // [256] col sums + sumsq
  float* prm  = sums + 2 * D_IN;                 // [256] scale + shift

  // 1) z = spmm(x)
  (void)hipMemsetAsync(z, 0, featBytes, stream);
  spmm_scatter<<<(E + 7) / 8, 256, 0, stream>>>(rows, cols, vals, x, z, E);

  // 2) h = z @ W1 + b1   (fp32 WMMA)
  gemm1_bias<<<N / 16, 256, 0, stream>>>(z, W1, b1, h);

  // 3) BatchNorm stats + apply + ReLU
  (void)hipMemsetAsync(sums, 0, 2 * D_IN * sizeof(float), stream);
  const int rowsPerBlk = 200;
  col_stats<<<(N + rowsPerBlk - 1) / rowsPerBlk, 128, 0, stream>>>(h, sums, N, rowsPerBlk);
  bn_params<<<1, 128, 0, stream>>>(sums, gamma1, beta1, prm, 1.0f / (float)N);
  int n4 = N * D_IN / 4;
  bn_relu<<<(n4 + 255) / 256, 256, 0, stream>>>(h, prm, n4);

  // 4) z = spmm(h)
  (void)hipMemsetAsync(z, 0, featBytes, stream);
  spmm_scatter<<<(E + 7) / 8, 256, 0, stream>>>(rows, cols, vals, h, z, E);

  // 5) out = log_softmax(z @ Wc + bc)  (fp32 WMMA + fused row reduction)
  gemm2_lsm<<<(N + 127) / 128, 256, 0, stream>>>(z, Wc, bc, out, N);
}